// ConjugatedGrad_49538152792773
// MI455X (gfx1250) — compile-verified
//
#include <hip/hip_runtime.h>

#define HDIM 320
#define HW   (HDIM * HDIM)        // 102400
#define BATCH 64
#define NITER 10
#define SHIFT 160                 // fftshift == ifftshift for even N

typedef __attribute__((ext_vector_type(2))) float v2f;
typedef __attribute__((ext_vector_type(8))) float v8f;
typedef __attribute__((ext_vector_type(4))) int   v4i;

// Async global->LDS copy path (gfx1250). Guarded: falls back to VALU staging
// if this toolchain does not declare the builtins.
#if defined(__has_builtin)
#if __has_builtin(__builtin_amdgcn_global_load_async_to_lds_b128) && \
    __has_builtin(__builtin_amdgcn_s_wait_asynccnt)
#define USE_ASYNC_LDS 1
#endif
#endif

__device__ __forceinline__ v8f wmma_f32(v2f a, v2f b, v8f c) {
    // D = A(16x4) * B(4x16) + C(16x16), fp32 WMMA (CDNA5)
    return __builtin_amdgcn_wmma_f32_16x16x4_f32(
        /*neg_a=*/false, a, /*neg_b=*/false, b,
        /*c_mod=*/(short)0, c, /*reuse_a=*/false, /*reuse_b=*/false);
}

// ---------------------------------------------------------------------------
// Build forward DFT matrix W (symmetric) and inverse V = conj(W)/320.
// Stored as [real plane | imag plane], each HW floats.
// ---------------------------------------------------------------------------
__global__ void build_dft_kernel(float* __restrict__ Wf, float* __restrict__ Vi) {
    int idx = blockIdx.x * blockDim.x + threadIdx.x;
    if (idx >= HW) return;
    int k = idx / HDIM, n = idx % HDIM;
    int kn = (k * n) % HDIM;  // reduce phase for accuracy
    float ang = -6.283185307179586f * (float)kn / (float)HDIM;
    float s, c;
    __sincosf(ang, &s, &c);
    Wf[idx]      = c;
    Wf[HW + idx] = s;
    const float inv = 1.0f / (float)HDIM;
    Vi[idx]      = c * inv;
    Vi[HW + idx] = -s * inv;
}

__global__ void zero_kernel(float* __restrict__ p, int n) {
    int i = blockIdx.x * blockDim.x + threadIdx.x;
    if (i < n) p[i] = 0.0f;
}

// ---------------------------------------------------------------------------
// Init: r = p = ifftshift(rhs), x = 0, maskS = ifftshift(mask), rTr += |r|^2
// grid (HW/256, BATCH)
// ---------------------------------------------------------------------------
__global__ void init_kernel(const float* __restrict__ rhs,
                            const float* __restrict__ mask,
                            float* __restrict__ r, float* __restrict__ p,
                            float* __restrict__ x, float* __restrict__ maskS,
                            float* __restrict__ rTr) {
    int b = blockIdx.y;
    int idx = blockIdx.x * blockDim.x + threadIdx.x;   // 0..HW-1
    int i = idx / HDIM, j = idx % HDIM;
    int si = (i + SHIFT) % HDIM, sj = (j + SHIFT) % HDIM;
    int sidx = si * HDIM + sj;

    size_t base = (size_t)b * 2 * HW;
    float vr = rhs[base + sidx];
    float vi = rhs[base + HW + sidx];

    r[base + idx] = vr;      r[base + HW + idx] = vi;
    p[base + idx] = vr;      p[base + HW + idx] = vi;
    x[base + idx] = 0.0f;    x[base + HW + idx] = 0.0f;
    maskS[(size_t)b * HW + idx] = mask[(size_t)b * HW + sidx];

    __shared__ float sred[256];
    sred[threadIdx.x] = vr * vr + vi * vi;
    __syncthreads();
    for (int s = 128; s > 0; s >>= 1) {
        if (threadIdx.x < s) sred[threadIdx.x] += sred[threadIdx.x + s];
        __syncthreads();
    }
    if (threadIdx.x == 0) atomicAdd(&rTr[b], sred[0]);
}

// ---------------------------------------------------------------------------
// Batched complex GEMM via fp32 WMMA:  C[b] = A[b] * B[b]  (320x320x320)
//   planes: real at ptr, imag at ptr+HW; per-batch stride in floats (0 = shared)
//   maskB  : optional per-batch HW mask applied multiplicatively to B loads
//   addP   : optional epilogue  C += (*lamPtr) * addP[b]
// Block: 256 threads = 8 waves; block tile 64x32; wave tile 16x16.
// A staged row-major (stride 20 floats -> 16B-aligned rows for b128 async);
// B staged TRANSPOSED (sBt[col][k], stride 18) so every WMMA fragment is one
// aligned ds_load_b64 straight into the operand register pair.
// grid (5, 10, BATCH)
// ---------------------------------------------------------------------------
#define BM 64
#define BN 32
#define BK 16
#define AK (BK + 4)   // 20 floats = 80 B row stride (16 B aligned)
#define BKS (BK + 2)  // 18 floats = 72 B row stride (8 B aligned)

__global__ __launch_bounds__(256)
void cgemm_wmma_kernel(const float* __restrict__ A, long long aStride,
                       const float* __restrict__ Bm, long long bStride,
                       float* __restrict__ C, long long cStride,
                       const float* __restrict__ maskB,
                       const float* __restrict__ addP,
                       const float* __restrict__ lamPtr) {
    __shared__ float sAr[BM][AK],  sAi[BM][AK];
    __shared__ float sBtr[BN][BKS], sBti[BN][BKS];

    const int b    = blockIdx.z;
    const int row0 = blockIdx.x * BM;
    const int col0 = blockIdx.y * BN;

    const float* Ar = A + (size_t)b * (size_t)aStride;
    const float* Ai = Ar + HW;
    const float* Br = Bm + (size_t)b * (size_t)bStride;
    const float* Bi = Br + HW;
    const float* Mk = maskB ? (maskB + (size_t)b * HW) : nullptr;

    const int t    = threadIdx.x;
    const int lane = t & 31;
    const int wid  = t >> 5;       // 0..7
    const int wm   = wid & 3;      // 0..3  (M tiles)
    const int wn   = wid >> 2;     // 0..1  (N tiles)
    const int half = (lane >> 4) & 1;
    const int lrow = lane & 15;
    const int mRow = wm * 16 + lrow;   // LDS row of this lane's A fragment
    const int nCol = wn * 16 + lrow;   // LDS row (=column) of B^T fragment

    // per-thread staging coordinates
    const int arr  = t >> 2;           // 0..63  A row
    const int akk  = (t & 3) * 4;      // 0,4,8,12  A col (4-float segment)

    v8f acc_rr = {}; v8f acc_ii = {}; v8f acc_ri = {}; v8f acc_ir = {};

    for (int kc = 0; kc < HDIM; kc += BK) {
        // ---- stage A strip (BM x BK): pure copy -> async LDS pipe if present
        {
            size_t gi = (size_t)(row0 + arr) * HDIM + kc + akk;
#if defined(USE_ASYNC_LDS)
            __builtin_amdgcn_global_load_async_to_lds_b128(
                (__attribute__((address_space(1))) v4i*)(Ar + gi),
                (__attribute__((address_space(3))) v4i*)&sAr[arr][akk], 0, 0);
            __builtin_amdgcn_global_load_async_to_lds_b128(
                (__attribute__((address_space(1))) v4i*)(Ai + gi),
                (__attribute__((address_space(3))) v4i*)&sAi[arr][akk], 0, 0);
#else
            #pragma unroll
            for (int u = 0; u < 4; ++u) {
                sAr[arr][akk + u] = Ar[gi + u];
                sAi[arr][akk + u] = Ai[gi + u];
            }
#endif
        }
        // ---- stage B strip (BK x BN) transposed, fused mask (VALU path)
        #pragma unroll
        for (int u = 0; u < 2; ++u) {
            int flat = t + u * 256;
            int kk = flat >> 5;        // 0..15
            int cc = flat & 31;        // 0..31
            size_t gi = (size_t)(kc + kk) * HDIM + col0 + cc;
            float m = Mk ? Mk[gi] : 1.0f;
            sBtr[cc][kk] = Br[gi] * m;
            sBti[cc][kk] = Bi[gi] * m;
        }
#if defined(USE_ASYNC_LDS)
        __builtin_amdgcn_s_wait_asynccnt(0);
#endif
        __syncthreads();

        // ---- prefetch next chunk's lines into cache (global_prefetch_b8)
        if (kc + BK < HDIM) {
            __builtin_prefetch(&Ar[(size_t)(row0 + arr) * HDIM + kc + BK + akk], 0, 0);
            __builtin_prefetch(&Ai[(size_t)(row0 + arr) * HDIM + kc + BK + akk], 0, 0);
            int kk = t >> 5, cc = (t & 31);
            __builtin_prefetch(&Br[(size_t)(kc + BK + kk) * HDIM + col0 + cc], 0, 0);
            __builtin_prefetch(&Bi[(size_t)(kc + BK + kk) * HDIM + col0 + cc], 0, 0);
        }

        #pragma unroll
        for (int ks = 0; ks < BK; ks += 4) {
            // fp32 16x16x4 fragment layout: lanes 0-15 hold K{0,1},
            // lanes 16-31 hold K{2,3} -> adjacent pair, one ds_load_b64 each.
            int k0 = ks + 2 * half;
            v2f ar = *(const v2f*)&sAr[mRow][k0];
            v2f ai = *(const v2f*)&sAi[mRow][k0];
            v2f br = *(const v2f*)&sBtr[nCol][k0];
            v2f bi = *(const v2f*)&sBti[nCol][k0];

            acc_rr = wmma_f32(ar, br, acc_rr);
            acc_ii = wmma_f32(ai, bi, acc_ii);
            acc_ri = wmma_f32(ar, bi, acc_ri);
            acc_ir = wmma_f32(ai, br, acc_ir);
        }
        __syncthreads();
    }

    float* Cr = C + (size_t)b * (size_t)cStride;
    float* Ci = Cr + HW;
    const float lam = lamPtr ? *lamPtr : 0.0f;
    const float* Pr = addP ? (addP + (size_t)b * 2 * HW) : nullptr;

    #pragma unroll
    for (int v = 0; v < 8; ++v) {
        int row = row0 + wm * 16 + v + 8 * half;  // C layout: VGPR v, M = v + 8*half
        int col = col0 + wn * 16 + lrow;          //           N = lane % 16
        size_t gi = (size_t)row * HDIM + col;
        float cr = acc_rr[v] - acc_ii[v];
        float ci = acc_ri[v] + acc_ir[v];
        if (Pr) {
            cr += lam * Pr[gi];
            ci += lam * Pr[HW + gi];
        }
        Cr[gi] = cr;
        Ci[gi] = ci;
    }
}

// ---------------------------------------------------------------------------
// pAp[b] += sum Re(conj(p) * Ap).   grid (HW/256, BATCH)
// ---------------------------------------------------------------------------
__global__ void reduce_pAp_kernel(const float* __restrict__ p,
                                  const float* __restrict__ Ap,
                                  float* __restrict__ pAp) {
    int b = blockIdx.y;
    int idx = blockIdx.x * blockDim.x + threadIdx.x;
    size_t base = (size_t)b * 2 * HW;
    float v = p[base + idx] * Ap[base + idx] + p[base + HW + idx] * Ap[base + HW + idx];
    __shared__ float sred[256];
    sred[threadIdx.x] = v;
    __syncthreads();
    for (int s = 128; s > 0; s >>= 1) {
        if (threadIdx.x < s) sred[threadIdx.x] += sred[threadIdx.x + s];
        __syncthreads();
    }
    if (threadIdx.x == 0) atomicAdd(&pAp[b], sred[0]);
}

// ---------------------------------------------------------------------------
// alpha = rTr/pAp ; x += alpha p ; r -= alpha Ap ; rTrNew += |r|^2
// ---------------------------------------------------------------------------
__global__ void update_xr_kernel(float* __restrict__ x, float* __restrict__ r,
                                 const float* __restrict__ p, const float* __restrict__ Ap,
                                 const float* __restrict__ rTr, const float* __restrict__ pAp,
                                 float* __restrict__ rTrNew) {
    int b = blockIdx.y;
    int idx = blockIdx.x * blockDim.x + threadIdx.x;
    size_t base = (size_t)b * 2 * HW;
    float alpha = rTr[b] / pAp[b];

    x[base + idx]      += alpha * p[base + idx];
    x[base + HW + idx] += alpha * p[base + HW + idx];
    float nr = r[base + idx]      - alpha * Ap[base + idx];
    float ni = r[base + HW + idx] - alpha * Ap[base + HW + idx];
    r[base + idx]      = nr;
    r[base + HW + idx] = ni;

    __shared__ float sred[256];
    sred[threadIdx.x] = nr * nr + ni * ni;
    __syncthreads();
    for (int s = 128; s > 0; s >>= 1) {
        if (threadIdx.x < s) sred[threadIdx.x] += sred[threadIdx.x + s];
        __syncthreads();
    }
    if (threadIdx.x == 0) atomicAdd(&rTrNew[b], sred[0]);
}

// beta = rTrNew/rTr ; p = r + beta p
__global__ void update_p_kernel(float* __restrict__ p, const float* __restrict__ r,
                                const float* __restrict__ rTr,
                                const float* __restrict__ rTrNew) {
    int b = blockIdx.y;
    int idx = blockIdx.x * blockDim.x + threadIdx.x;
    size_t base = (size_t)b * 2 * HW;
    float beta = rTrNew[b] / rTr[b];
    p[base + idx]      = r[base + idx]      + beta * p[base + idx];
    p[base + HW + idx] = r[base + HW + idx] + beta * p[base + HW + idx];
}

__global__ void commit_rtr_kernel(float* __restrict__ rTr, const float* __restrict__ rTrNew) {
    int b = threadIdx.x;
    if (b < BATCH) rTr[b] = rTrNew[b];
}

// out[b,i,j,{0,1}] = fftshift(x)  (interleave planes, apply final shift)
__global__ void final_out_kernel(const float* __restrict__ x, float* __restrict__ out) {
    int b = blockIdx.y;
    int idx = blockIdx.x * blockDim.x + threadIdx.x;
    int i = idx / HDIM, j = idx % HDIM;
    int si = (i + SHIFT) % HDIM, sj = (j + SHIFT) % HDIM;
    size_t s = (size_t)b * 2 * HW + (size_t)si * HDIM + sj;
    size_t o = ((size_t)b * HW + idx) * 2;
    out[o]     = x[s];
    out[o + 1] = x[s + HW];
}

// ---------------------------------------------------------------------------
extern "C" void kernel_launch(void* const* d_in, const int* in_sizes, int n_in,
                              void* d_out, int out_size, void* d_ws, size_t ws_size,
                              hipStream_t stream) {
    const float* rhs    = (const float*)d_in[0];  // [B,2,H,W]
    const float* mask   = (const float*)d_in[1];  // [B,H,W]
    const float* lamPtr = (const float*)d_in[2];  // device scalar

    float* ws = (float*)d_ws;
    const size_t PB = (size_t)BATCH * 2 * HW;     // one complex batch buffer
    float* Wf     = ws;                 // 2*HW
    float* Vi     = Wf + 2 * HW;        // 2*HW
    float* maskS  = Vi + 2 * HW;        // B*HW
    float* x      = maskS + (size_t)BATCH * HW;
    float* r      = x  + PB;
    float* p      = r  + PB;
    float* Ap     = p  + PB;
    float* T1     = Ap + PB;
    float* T2     = T1 + PB;
    float* rTr    = T2 + PB;            // 64
    float* rTrNew = rTr + BATCH;        // 64
    float* pAp    = rTrNew + BATCH;     // 64

    const dim3 blk(256);
    const dim3 gridEw(HW / 256, BATCH);            // elementwise per-batch
    const dim3 gridGemm(HDIM / BM, HDIM / BN, BATCH);

    // DFT matrices + CG init
    build_dft_kernel<<<dim3((HW + 255) / 256), blk, 0, stream>>>(Wf, Vi);
    zero_kernel<<<dim3(1), dim3(192), 0, stream>>>(rTr, 3 * BATCH);
    init_kernel<<<gridEw, blk, 0, stream>>>(rhs, mask, r, p, x, maskS, rTr);

    const long long S = 2 * HW;  // per-batch plane-pair stride
    for (int it = 0; it < NITER; ++it) {
        // AtA(p): T1 = W p ; T2 = T1 W (W symmetric) ; T1 = V (T2 .* mask) ;
        //         Ap = T1 V + lam p
        cgemm_wmma_kernel<<<gridGemm, blk, 0, stream>>>(Wf, 0, p, S, T1, S,
                                                        nullptr, nullptr, nullptr);
        cgemm_wmma_kernel<<<gridGemm, blk, 0, stream>>>(T1, S, Wf, 0, T2, S,
                                                        nullptr, nullptr, nullptr);
        cgemm_wmma_kernel<<<gridGemm, blk, 0, stream>>>(Vi, 0, T2, S, T1, S,
                                                        maskS, nullptr, nullptr);
        cgemm_wmma_kernel<<<gridGemm, blk, 0, stream>>>(T1, S, Vi, 0, Ap, S,
                                                        nullptr, p, lamPtr);

        zero_kernel<<<dim3(1), dim3(128), 0, stream>>>(rTrNew, 2 * BATCH);
        reduce_pAp_kernel<<<gridEw, blk, 0, stream>>>(p, Ap, pAp);
        update_xr_kernel<<<gridEw, blk, 0, stream>>>(x, r, p, Ap, rTr, pAp, rTrNew);
        update_p_kernel<<<gridEw, blk, 0, stream>>>(p, r, rTr, rTrNew);
        commit_rtr_kernel<<<dim3(1), dim3(64), 0, stream>>>(rTr, rTrNew);
    }

    final_out_kernel<<<gridEw, blk, 0, stream>>>(x, (float*)d_out);
}